// SequenceLabeling_62423054680396
// MI455X (gfx1250) — compile-verified
//
#include <hip/hip_runtime.h>

typedef __attribute__((ext_vector_type(2))) float v2f;
typedef __attribute__((ext_vector_type(8))) float v8f;

#define HID   1024
#define CLS   32
#define SEQ   512
#define BATCH 64
#define MROWS (BATCH * SEQ)   // 32768 flattened rows

// ---------------------------------------------------------------------------
// Kernel 1: scores = relu(hidden @ W + b)   via V_WMMA_F32_16X16X4_F32
//   hidden: [32768, 1024] row-major, W: [1024, 32], b: [32]
//   One wave -> one 16x32 output tile (two 16x16 N-tiles), K accumulated in
//   256 steps of k=4. A-frag layout (ISA 7.12.2, 32-bit A 16x4):
//     lanes 0-15:  M=lane,    VGPR0=K+0, VGPR1=K+1
//     lanes 16-31: M=lane-16, VGPR0=K+2, VGPR1=K+3
//   B-frag (4x16) mirrors with lane -> N. D: VGPR v -> M = v + 8*(lane>=16),
//   N = lane&15.
// ---------------------------------------------------------------------------
__global__ __launch_bounds__(256)
void emis_gemm_relu(const float* __restrict__ hidden,
                    const float* __restrict__ W,
                    const float* __restrict__ bias,
                    float* __restrict__ scores)
{
    const int lane = threadIdx.x & 31;
    const int wave = threadIdx.x >> 5;
    const int tile = blockIdx.x * 8 + wave;     // 0..2047
    const int row0 = tile << 4;                 // 16 rows per tile
    const int h    = lane >> 4;                 // half-wave select (0/1)
    const int lid  = lane & 15;

    // A: lane reads hidden[row0+lid][k + 2h .. k + 2h + 1] each k-step (8B load)
    const float* aptr = hidden + (size_t)(row0 + lid) * HID + (h << 1);
    // B: lane reads W[k + 2h][lid] / W[k + 2h + 1][lid] (L2-resident, 128 KB)
    const float* bptr = W + (size_t)(h << 1) * CLS + lid;

    v8f c0 = {};   // N-tile 0 (cols 0..15)
    v8f c1 = {};   // N-tile 1 (cols 16..31)

    for (int k = 0; k < HID; k += 4) {
        v2f a = *(const v2f*)(aptr + k);
        v2f b0, b1;
        b0.x = bptr[(k + 0) * CLS];
        b0.y = bptr[(k + 1) * CLS];
        b1.x = bptr[(k + 0) * CLS + 16];
        b1.y = bptr[(k + 1) * CLS + 16];
        c0 = __builtin_amdgcn_wmma_f32_16x16x4_f32(false, a, false, b0,
                                                   (short)0, c0, false, false);
        c1 = __builtin_amdgcn_wmma_f32_16x16x4_f32(false, a, false, b1,
                                                   (short)0, c1, false, false);
    }

    const float bb0 = bias[lid];
    const float bb1 = bias[lid + 16];
#pragma unroll
    for (int v = 0; v < 8; ++v) {
        const int row = row0 + v + (h << 3);
        const float r0 = fmaxf(c0[v] + bb0, 0.0f);
        const float r1 = fmaxf(c1[v] + bb1, 0.0f);
        scores[(size_t)row * CLS + lid]      = r0;
        scores[(size_t)row * CLS + lid + 16] = r1;
    }
}

// ---------------------------------------------------------------------------
// Kernel 2: Viterbi. One wave32 per batch element; lane j = state j.
//   trellis[t][j] = max_i trellis[t-1][i] * T[i][j] * s[t][j]
//   prev[i] broadcast via v_readlane (constant lane after unroll).
//   Backpointers in LDS (512*32 bytes); lane 0 backtracks serially.
// ---------------------------------------------------------------------------
__device__ __forceinline__ float lane_bcast(float v, int srclane) {
    return __builtin_bit_cast(float,
        __builtin_amdgcn_readlane(__builtin_bit_cast(int, v), srclane));
}

__global__ __launch_bounds__(32)
void viterbi_kernel(const float* __restrict__ scores,   // [64,512,32]
                    const float* __restrict__ T,        // [32,32]
                    int*  __restrict__ paths,           // [64,512]
                    float* __restrict__ trellis)        // [64,512,32]
{
    __shared__ unsigned char bp[SEQ * CLS];   // backpointers, rows 1..511 used

    const int b = blockIdx.x;
    const int j = threadIdx.x;                // state index == lane

    // T column j into registers (32 VGPRs)
    float tc[CLS];
#pragma unroll
    for (int i = 0; i < CLS; ++i) tc[i] = T[i * CLS + j];

    const float* sb = scores  + (size_t)b * SEQ * CLS;
    float*       tb = trellis + (size_t)b * SEQ * CLS;

    float prev = sb[j];
    tb[j] = prev;

    for (int t = 1; t < SEQ; ++t) {
        const float s = sb[t * CLS + j];
        float best = -__builtin_inff();
        int   bi   = 0;
#pragma unroll
        for (int i = 0; i < CLS; ++i) {
            const float pv = lane_bcast(prev, i);
            const float v  = pv * tc[i] * s;
            const bool  g  = v > best;       // strict ">" -> first-index ties
            best = g ? v : best;
            bi   = g ? i : bi;
        }
        prev = best;
        tb[t * CLS + j]  = best;
        bp[t * CLS + j]  = (unsigned char)bi;
    }

    // argmax over final trellis row (first max wins), computed uniformly
    float fbest = lane_bcast(prev, 0);
    int   fbi   = 0;
#pragma unroll
    for (int i = 1; i < CLS; ++i) {
        const float pv = lane_bcast(prev, i);
        const bool  g  = pv > fbest;
        fbest = g ? pv : fbest;
        fbi   = g ? i : fbi;
    }

    __syncthreads();   // single-wave WG: cheap; orders LDS writes before scan

    if (j == 0) {
        int* pb = paths + (size_t)b * SEQ;
        int cur = fbi;
        pb[SEQ - 1] = cur;
        for (int t = SEQ - 1; t >= 1; --t) {
            cur = (int)bp[t * CLS + cur];
            pb[t - 1] = cur;
        }
    }
}

// ---------------------------------------------------------------------------
extern "C" void kernel_launch(void* const* d_in, const int* in_sizes, int n_in,
                              void* d_out, int out_size, void* d_ws, size_t ws_size,
                              hipStream_t stream)
{
    const float* hidden = (const float*)d_in[0];   // [64,512,1024]
    const float* W      = (const float*)d_in[1];   // [1024,32]
    const float* bias   = (const float*)d_in[2];   // [32]
    const float* T      = (const float*)d_in[3];   // [32,32]

    float* scores  = (float*)d_ws;                 // [32768,32] = 4 MB scratch

    // d_out = paths (int32, 64*512) followed by trellis (f32, 64*512*32)
    int*   paths   = (int*)d_out;
    float* trellis = (float*)d_out + (size_t)BATCH * SEQ;

    // 2048 tiles of 16 rows, 8 waves (tiles) per 256-thread block
    emis_gemm_relu<<<dim3(MROWS / 16 / 8), dim3(256), 0, stream>>>(
        hidden, W, bias, scores);

    viterbi_kernel<<<dim3(BATCH), dim3(32), 0, stream>>>(
        scores, T, paths, trellis);
}